// GNNIDS_51737176047725
// MI455X (gfx1250) — compile-verified
//
#include <hip/hip_runtime.h>
#include <hip/hip_bf16.h>

typedef __attribute__((ext_vector_type(2))) float v2f;
typedef __attribute__((ext_vector_type(4))) float v4f;
typedef __attribute__((ext_vector_type(8))) float v8f;

// ---------------- utility kernels ----------------

__global__ void k_fill(float* __restrict__ p, float v, int n) {
  int i = blockIdx.x * blockDim.x + threadIdx.x;
  if (i < n) p[i] = v;
}

__global__ void k_deg_accum(const int* __restrict__ dst, float* __restrict__ deg, int E) {
  int e = blockIdx.x * blockDim.x + threadIdx.x;
  if (e < E) atomicAdd(&deg[dst[e]], 1.0f);
}

__global__ void k_rsqrt_inplace(float* __restrict__ p, int n) {
  int i = blockIdx.x * blockDim.x + threadIdx.x;
  if (i < n) p[i] = rsqrtf(p[i]);  // deg >= 1 guaranteed by self-loop init
}

// pack x [N x 15] -> xp [Npad x 16], zero padded (cols 15, rows >= N)
__global__ void k_pack_x(const float* __restrict__ x, float* __restrict__ xp,
                         int N, int Npad) {
  int idx = blockIdx.x * blockDim.x + threadIdx.x;
  if (idx >= Npad * 16) return;
  int node = idx >> 4;
  int j = idx & 15;
  xp[idx] = (node < N && j < 15) ? x[node * 15 + j] : 0.0f;
}

// pack W [K x outd] -> Wp [16 x 16], zero padded. one block of 256 threads.
__global__ void k_pack_w(const float* __restrict__ W, float* __restrict__ Wp,
                         int K, int outd) {
  int idx = threadIdx.x;  // 0..255
  int r = idx >> 4;
  int cc = idx & 15;
  Wp[idx] = (r < K && cc < outd) ? W[r * outd + cc] : 0.0f;
}

// ---------------- dense transform: HW = H @ Wp via V_WMMA_F32_16X16X4_F32 --
// Fully padded: H is [Npad x 16] (stride 16, zero-padded), Wp is [16 x 16].
// One wave32 per 16-row tile; branch-free inner body.
// A frag (16x4 f32): lanes 0-15 hold K=k0,k0+1 ; lanes 16-31 hold K=k0+2,k0+3.
// C/D (16x16 f32): VGPR v -> row m0+v (lanes 0-15) / m0+v+8 (lanes 16-31), col = lane&15.

__global__ void k_gemm_wmma(const float* __restrict__ H, const float* __restrict__ Wp,
                            float* __restrict__ HW, int nTiles) {
  int wave = (int)((blockIdx.x * blockDim.x + threadIdx.x) >> 5);
  if (wave >= nTiles) return;          // whole-wave exit keeps EXEC all-ones
  int lane = (int)(threadIdx.x & 31);
  int half = lane >> 4;                // 0: lanes 0-15, 1: lanes 16-31
  int lm = lane & 15;
  int m0 = wave * 16;

  const float* hrow = H + (size_t)(m0 + lm) * 16 + half * 2;

  // hoist all four B fragments (W is 16x16, resident in L2/registers)
  v2f b0, b1, b2, b3;
  {
    int kr = half * 2;
    b0.x = Wp[(kr + 0) * 16 + lm];  b0.y = Wp[(kr + 1) * 16 + lm];
    b1.x = Wp[(kr + 4) * 16 + lm];  b1.y = Wp[(kr + 5) * 16 + lm];
    b2.x = Wp[(kr + 8) * 16 + lm];  b2.y = Wp[(kr + 9) * 16 + lm];
    b3.x = Wp[(kr + 12) * 16 + lm]; b3.y = Wp[(kr + 13) * 16 + lm];
  }

  v2f a0 = *(const v2f*)(hrow + 0);
  v2f a1 = *(const v2f*)(hrow + 4);
  v2f a2 = *(const v2f*)(hrow + 8);
  v2f a3 = *(const v2f*)(hrow + 12);

  v8f c = {};
  c = __builtin_amdgcn_wmma_f32_16x16x4_f32(false, a0, false, b0, (short)0, c, false, false);
  c = __builtin_amdgcn_wmma_f32_16x16x4_f32(false, a1, false, b1, (short)0, c, false, false);
  c = __builtin_amdgcn_wmma_f32_16x16x4_f32(false, a2, false, b2, (short)0, c, false, false);
  c = __builtin_amdgcn_wmma_f32_16x16x4_f32(false, a3, false, b3, (short)0, c, false, false);

#pragma unroll
  for (int v = 0; v < 8; ++v) {
    HW[(size_t)(m0 + v + half * 8) * 16 + lm] = c[v];
  }
}

// ---------------- sparse aggregation -------------------------------------
// acc[dst] += hw[src] * dinv[src]*dinv[dst]; compile-time OUTD -> wide gathers.

template <int OUTD>
__global__ void k_scatter(const int* __restrict__ src, const int* __restrict__ dst,
                          const float* __restrict__ dinv, const float* __restrict__ hw,
                          float* __restrict__ acc, int E) {
  int e = blockIdx.x * blockDim.x + threadIdx.x;
  if (e >= E) return;
  int s = src[e];
  int d = dst[e];
  float w = dinv[s] * dinv[d];
  const float* hs = hw + (size_t)s * 16;
  float* ad = acc + (size_t)d * 16;
  __builtin_prefetch(hs, 0, 1);  // global_prefetch_b8
#pragma unroll
  for (int j = 0; j < OUTD; ++j) {
    atomicAdd(&ad[j], hs[j] * w);
  }
}

// out[node,0..15] = j<outd ? act(acc + hw*dinv^2 + bias[j]) : 0  (keeps padding)
__global__ void k_finalize(const float* __restrict__ acc, const float* __restrict__ hw,
                           const float* __restrict__ dinv, const float* __restrict__ bias,
                           float* __restrict__ out, int N, int outd, int do_relu) {
  int idx = blockIdx.x * blockDim.x + threadIdx.x;
  if (idx >= N * 16) return;
  int node = idx >> 4;
  int j = idx & 15;
  float r = 0.0f;
  if (j < outd) {
    float di = dinv[node];
    float v = acc[idx] + hw[idx] * di * di + bias[j];
    r = do_relu ? fmaxf(v, 0.0f) : v;
  }
  out[idx] = r;
}

// anomaly score: mean over 15 features of (x - recon)^2 ; recon stride 16
__global__ void k_score(const float* __restrict__ x, const float* __restrict__ recon,
                        float* __restrict__ score, int N) {
  int i = blockIdx.x * blockDim.x + threadIdx.x;
  if (i >= N) return;
  const float* xi = x + (size_t)i * 15;
  const float* ri = recon + (size_t)i * 16;
  float s = 0.0f;
#pragma unroll
  for (int j = 0; j < 15; ++j) {
    float d = xi[j] - ri[j];
    s += d * d;
  }
  score[i] = s * (1.0f / 15.0f);
}

// ---------------- driver ---------------------------------------------------

extern "C" void kernel_launch(void* const* d_in, const int* in_sizes, int n_in,
                              void* d_out, int out_size, void* d_ws, size_t ws_size,
                              hipStream_t stream) {
  const float* x     = (const float*)d_in[0];
  const int*   ei    = (const int*)d_in[1];
  const float* encW1 = (const float*)d_in[2];
  const float* encb1 = (const float*)d_in[3];
  const float* encW2 = (const float*)d_in[4];
  const float* encb2 = (const float*)d_in[5];
  const float* encW3 = (const float*)d_in[6];
  const float* encb3 = (const float*)d_in[7];
  const float* decW1 = (const float*)d_in[8];
  const float* decb1 = (const float*)d_in[9];
  const float* decW2 = (const float*)d_in[10];
  const float* decb2 = (const float*)d_in[11];
  const float* decW3 = (const float*)d_in[12];
  const float* decb3 = (const float*)d_in[13];

  const int N = in_sizes[0] / 15;
  const int E = in_sizes[1] / 2;
  const int Npad = (N + 15) & ~15;
  const int nTiles = Npad / 16;
  const int* srcp = ei;      // edge_index[0]
  const int* dstp = ei + E;  // edge_index[1]

  // workspace partition (all 64B aligned: Npad multiples of 16 floats)
  float* dinv = (float*)d_ws;                 // [Npad]
  float* wp   = dinv + Npad;                  // [256] padded weight 16x16
  float* hw   = wp + 256;                     // [Npad*16]
  float* acc  = hw + (size_t)Npad * 16;       // [Npad*16]
  float* hA   = acc + (size_t)Npad * 16;      // [Npad*16]
  float* hB   = hA + (size_t)Npad * 16;       // [Npad*16]
  float* xp   = hB + (size_t)Npad * 16;       // [Npad*16]

  const int T = 256;

  // symmetric normalization: deg (with self loop) -> dinv = rsqrt(deg)
  k_fill<<<(N + T - 1) / T, T, 0, stream>>>(dinv, 1.0f, N);
  k_deg_accum<<<(E + T - 1) / T, T, 0, stream>>>(dstp, dinv, E);
  k_rsqrt_inplace<<<(N + T - 1) / T, T, 0, stream>>>(dinv, N);

  // pad input features
  k_pack_x<<<((Npad * 16) + T - 1) / T, T, 0, stream>>>(x, xp, N, Npad);

  int gemmBlocks = (nTiles * 32 + T - 1) / T;

  auto layer = [&](const float* hp, int indim, const float* W, const float* b,
                   int outd, float* outp, int relu) {
    k_pack_w<<<1, 256, 0, stream>>>(W, wp, indim, outd);
    k_gemm_wmma<<<gemmBlocks, T, 0, stream>>>(hp, wp, hw, nTiles);
    k_fill<<<((Npad * 16) + T - 1) / T, T, 0, stream>>>(acc, 0.0f, Npad * 16);
    if (outd == 8) {
      k_scatter<8><<<(E + T - 1) / T, T, 0, stream>>>(srcp, dstp, dinv, hw, acc, E);
    } else if (outd == 2) {
      k_scatter<2><<<(E + T - 1) / T, T, 0, stream>>>(srcp, dstp, dinv, hw, acc, E);
    } else {
      k_scatter<15><<<(E + T - 1) / T, T, 0, stream>>>(srcp, dstp, dinv, hw, acc, E);
    }
    k_finalize<<<((N * 16) + T - 1) / T, T, 0, stream>>>(acc, hw, dinv, b, outp, N, outd, relu);
    if (Npad > N) {  // keep pad rows zero for next layer's GEMM
      k_fill<<<(((Npad - N) * 16) + T - 1) / T, T, 0, stream>>>(
          outp + (size_t)N * 16, 0.0f, (Npad - N) * 16);
    }
  };

  // encoder
  layer(xp, 15, encW1, encb1, 8, hA, 1);
  layer(hA,  8, encW2, encb2, 8, hB, 1);
  layer(hB,  8, encW3, encb3, 2, hA, 0);   // latent, no relu
  // decoder
  layer(hA,  2, decW1, decb1, 8,  hB, 1);
  layer(hB,  8, decW2, decb2, 8,  hA, 1);
  layer(hA,  8, decW3, decb3, 15, hB, 0);  // recon, no relu

  k_score<<<(N + T - 1) / T, T, 0, stream>>>(x, hB, (float*)d_out, N);
}